// HMPDCA_Dist_11278584119461
// MI455X (gfx1250) — compile-verified
//
#include <hip/hip_runtime.h>
#include <hip/hip_bf16.h>
#include <math.h>

// ---------------------------------------------------------------------------
// HMPDCA hyperbolic dilated attention, CDNA5 (gfx1250, wave32).
//
// One wave handles a (t, 16-row n-tile). Per-row dot products (x.x, y.y, x.y)
// are computed as diagonals of 16x16 fp32 WMMA products:
//     C = X * Y^T  via 16 chained V_WMMA_F32_16X16X4_F32 ops (K=64).
// For row-major fp32 tiles the per-lane A-layout and B-layout coincide
// (lane = row index, K pair split across lane halves), so a single float2
// load per lane per K-chunk feeds both operands.
// ---------------------------------------------------------------------------

typedef __attribute__((ext_vector_type(2))) float v2f;
typedef __attribute__((ext_vector_type(8))) float v8f;

#define T_DIM 12
#define N_DIM 30000
#define D_DIM 64
#define DIL   2
#define MIN_NORM   1e-15f
#define ATANH_CLIP 1e-7f

#define TILES_PER_T (N_DIM / 16)          // 1875
#define TOTAL_WAVES (T_DIM * TILES_PER_T) // 22500
#define WAVES_PER_BLOCK 4
#define BLOCKS (TOTAL_WAVES / WAVES_PER_BLOCK) // 5625 (exact)

// C/D 16x16 f32 layout: VGPR r, lanes 0-15 -> (M=r,   N=lane),
//                               lanes 16-31 -> (M=r+8, N=lane-16).
// Store as tile[N][M] so each lane writes 8 contiguous floats, then the
// diagonal element for row r is tile[r][r] (broadcast to both lanes of row r).
__device__ __forceinline__ float wave_diag(const v8f& C, float* tile, int lane)
{
    const int row  = lane & 15;
    const int colb = (lane >> 4) * 8;
#pragma unroll
    for (int i = 0; i < 8; ++i)
        tile[row * 16 + colb + i] = C[i];
    __syncthreads();
    float d = tile[row * 16 + row];
    __syncthreads();
    return d;
}

// C += A * B over K=64 in 16 WMMA steps of K=4.
__device__ __forceinline__ v8f wmma_chain(const v2f* A, const v2f* B)
{
    v8f C = {};
#pragma unroll
    for (int ch = 0; ch < 16; ++ch)
        C = __builtin_amdgcn_wmma_f32_16x16x4_f32(
                /*neg_a=*/false, A[ch], /*neg_b=*/false, B[ch],
                /*c_mod=*/(short)0, C, /*reuse_a=*/false, /*reuse_b=*/false);
    return C;
}

// lamb(z_k) * sigmoid(sqdist(x, y)) from the three dot products.
__device__ __forceinline__ float hyper_weight(float x2, float y2, float xy, float cc)
{
    // mobius_add(-x, y): num = A*(-x) + B*y
    float A    = 1.f - 2.f * cc * xy + cc * y2;
    float B    = 1.f - cc * x2;
    float num2 = fmaxf(A * A * x2 - 2.f * A * B * xy + B * B * y2, 0.f);
    float den  = fmaxf(1.f - 2.f * cc * xy + cc * cc * x2 * y2, MIN_NORM);
    float nrm  = sqrtf(num2) / den;                       // ||mobius_add||
    float sc   = sqrtf(cc);
    float u    = fminf(sc * nrm, 1.f - ATANH_CLIP);       // u >= 0
    float ath  = 0.5f * logf((1.f + u) / (1.f - u));      // artanh
    float dg   = ath * 2.f / sc;
    float kern = dg * dg;                                 // sqdist
    float i1   = 1.f / (1.f + cc * x2);
    float zn2  = 4.f * x2 * i1 * i1;                      // ||p2k(x)||^2
    float lamb = rsqrtf(fmaxf(1.f - cc * zn2, MIN_NORM)); // lorenz factor
    float sig  = 1.f / (1.f + expf(-kern));
    return lamb * sig;
}

__global__ void __launch_bounds__(128)
hmpdca_kernel(const float* __restrict__ hw, const float* __restrict__ cptr,
              float* __restrict__ out)
{
    __shared__ float tiles[WAVES_PER_BLOCK][16 * 16];

    const int tid  = threadIdx.x;
    const int lane = tid & 31;
    const int wblk = tid >> 5;
    float*    tile = tiles[wblk];

    const int wgid = blockIdx.x * WAVES_PER_BLOCK + wblk; // 0..22499
    const int t    = wgid / TILES_PER_T;
    const int n0   = (wgid % TILES_PER_T) * 16;

    const int r  = lane & 15;        // row (M and N index) this lane serves
    const int cb = (lane >> 4) * 2;  // K-pair offset within each 4-chunk
    const float cc = *cptr;

    // ---- load the three 16x64 tiles in WMMA A/B layout (float2 per chunk) --
    v2f y[16], x1[16], x2[16];
    {
        const float* p = hw + ((size_t)t * N_DIM + (n0 + r)) * D_DIM;
#pragma unroll
        for (int ch = 0; ch < 16; ++ch)
            y[ch] = *(const v2f*)(p + 4 * ch + cb);
    }
    const int t1 = t - DIL, t2 = t - 2 * DIL;  // wave-uniform branches
    if (t1 >= 0) {
        const float* p = hw + ((size_t)t1 * N_DIM + (n0 + r)) * D_DIM;
#pragma unroll
        for (int ch = 0; ch < 16; ++ch)
            x1[ch] = *(const v2f*)(p + 4 * ch + cb);
    } else {
#pragma unroll
        for (int ch = 0; ch < 16; ++ch) x1[ch] = (v2f)0.f;
    }
    if (t2 >= 0) {
        const float* p = hw + ((size_t)t2 * N_DIM + (n0 + r)) * D_DIM;
#pragma unroll
        for (int ch = 0; ch < 16; ++ch)
            x2[ch] = *(const v2f*)(p + 4 * ch + cb);
    } else {
#pragma unroll
        for (int ch = 0; ch < 16; ++ch) x2[ch] = (v2f)0.f;
    }

    // ---- per-row dot products via WMMA diagonals ---------------------------
    float y2  = wave_diag(wmma_chain(y,  y ), tile, lane); // y.y
    float x2a = wave_diag(wmma_chain(x1, x1), tile, lane); // x1.x1
    float xya = wave_diag(wmma_chain(x1, y ), tile, lane); // x1.y
    float x2b = wave_diag(wmma_chain(x2, x2), tile, lane); // x2.x2
    float xyb = wave_diag(wmma_chain(x2, y ), tile, lane); // x2.y

    // ---- per-row scalars (k=0 is y against itself: kernel==0) --------------
    float w0 = hyper_weight(y2,  y2, y2,  cc);
    float w1 = hyper_weight(x2a, y2, xya, cc);
    float w2 = hyper_weight(x2b, y2, xyb, cc);

    float wsum = fmaxf(w0 + w1 + w2, MIN_NORM);
    // z_k = 2*x_k/(1+c*x2_k); fold scale into g_k, divide by wsum once.
    float g0 = 2.f * w0 / ((1.f + cc * y2 ) * wsum);
    float g1 = 2.f * w1 / ((1.f + cc * x2a) * wsum);
    float g2 = 2.f * w2 / ((1.f + cc * x2b) * wsum);

    // ---- weighted Klein mean + its squared norm ----------------------------
    v2f zt[16];
    float ss = 0.f;
#pragma unroll
    for (int ch = 0; ch < 16; ++ch) {
        v2f z = g0 * y[ch] + g1 * x1[ch] + g2 * x2[ch];
        zt[ch] = z;
        ss += z.x * z.x + z.y * z.y;
    }
    ss += __shfl_xor(ss, 16, 32);  // partner lane holds other half of the row

    // ---- k2p and store -----------------------------------------------------
    float inv = 1.f / (1.f + sqrtf(fmaxf(1.f - cc * ss, MIN_NORM)));
    float* op = out + ((size_t)t * N_DIM + (n0 + r)) * D_DIM;
#pragma unroll
    for (int ch = 0; ch < 16; ++ch)
        *(v2f*)(op + 4 * ch + cb) = zt[ch] * inv;
}

extern "C" void kernel_launch(void* const* d_in, const int* in_sizes, int n_in,
                              void* d_out, int out_size, void* d_ws, size_t ws_size,
                              hipStream_t stream)
{
    (void)in_sizes; (void)n_in; (void)out_size; (void)d_ws; (void)ws_size;
    const float* hw = (const float*)d_in[0];   // hidden_window (T,N,D) f32
    const float* c_ = (const float*)d_in[1];   // scalar curvature
    float* out = (float*)d_out;                // (T,N,D) f32
    hmpdca_kernel<<<BLOCKS, WAVES_PER_BLOCK * 32, 0, stream>>>(hw, c_, out);
}